// GNNInvariant_38208029065918
// MI455X (gfx1250) — compile-verified
//
#include <hip/hip_runtime.h>
#include <hip/hip_bf16.h>

typedef __attribute__((ext_vector_type(2))) float v2f;
typedef __attribute__((ext_vector_type(8))) float v8f;

#define DSTATE 64
#define DIN    68          // STATE_DIM + 4
#define KPAIRS 34          // DIN / 2
#define NUM_ROUNDS 3       // reference constant (python scalar; fixed)
#define WPB 8              // waves per block

// ---------------------------------------------------------------------------
// Zero-fill f32
__global__ void gnn_zero_kernel(float* __restrict__ p, long n) {
    long i = (long)blockIdx.x * blockDim.x + threadIdx.x;
    if (i < n) p[i] = 0.0f;
}

// ---------------------------------------------------------------------------
// Edge-invariant features: [len, sum|c_from|, c_from.c_to, c_from.evec]
__global__ void gnn_feats_kernel(const float* __restrict__ coords,
                                 const float* __restrict__ elen,
                                 const float* __restrict__ evec,
                                 const int*   __restrict__ efrom,
                                 const int*   __restrict__ eto,
                                 float4* __restrict__ feats, int E) {
    int e = blockIdx.x * blockDim.x + threadIdx.x;
    if (e >= E) return;
    int f = efrom[e], t = eto[e];
    float fx = coords[3*f+0], fy = coords[3*f+1], fz = coords[3*f+2];
    float tx = coords[3*t+0], ty = coords[3*t+1], tz = coords[3*t+2];
    float vx = evec[3*e+0],  vy = evec[3*e+1],  vz = evec[3*e+2];
    float4 o;
    o.x = elen[e];
    o.y = fabsf(fx) + fabsf(fy) + fabsf(fz);
    o.z = fx*tx + fy*ty + fz*tz;
    o.w = fx*vx + fy*vy + fz*vz;
    feats[e] = o;
}

// ---------------------------------------------------------------------------
// Message passing round: one wave computes a 16-edge tile,
//   D[16,64] = tanh( [state[from] | feats] (16x68)  @  W (68x64) + b )
// via 17 K-steps x 4 N-tiles of v_wmma_f32_16x16x4_f32 (fully unrolled).
// W is staged in LDS K-pair-interleaved so every B fragment is a single
// aligned ds_load_b64 into a WMMA-ready even VGPR pair (no shuffle movs).
__global__ void __launch_bounds__(256)
gnn_msg_kernel(const float* __restrict__ state_prev,
               float*       __restrict__ state_next,
               const float4* __restrict__ feats,
               const float* __restrict__ Wmsg,   // [68,64] row-major
               const float* __restrict__ bmsg,   // [64]
               const int*   __restrict__ efrom,
               const int*   __restrict__ eto,
               int E) {
    // sW2[p][c][j] = W[2p + j][c]  (pair-interleaved along K)
    __shared__ float sW2[KPAIRS * DSTATE * 2];
    __shared__ float sB[DSTATE];                 // bias
    __shared__ float sA[WPB][16 * DIN];          // per-wave A staging (stride 68)

    const int tid = threadIdx.x;

    // Stage W into LDS in pair-interleaved layout.
    for (int i = tid; i < DIN * DSTATE; i += 256) {
        int k = i >> 6, c = i & 63;              // Wmsg[k][c]
        sW2[(k >> 1) * (DSTATE * 2) + c * 2 + (k & 1)] = Wmsg[i];
    }
    if (tid < DSTATE) sB[tid] = bmsg[tid];

    const int wave  = tid >> 5;
    const int lane  = tid & 31;
    const int row16 = lane & 15;       // M row within tile
    const int hi    = lane >> 4;       // half-wave selector (K pairs / M+8)
    const long tileBase = ((long)blockIdx.x * WPB + wave) * 16L;

    // Stage A rows: gather state[from] (64 f32 as 16 float4, split across
    // half-waves) plus feats (1 float4) into LDS.
    float* aw = &sA[wave][0];
    {
        long e  = tileBase + row16;
        long ec = (e < E) ? e : (long)(E - 1);
        int  src = efrom[ec];
        const float4* srow = (const float4*)(state_prev + (long)src * DSTATE);
        float4* drow = (float4*)(aw + row16 * DIN);   // 68*4=272B, 16B aligned
        #pragma unroll
        for (int j = 0; j < 8; ++j) drow[hi * 8 + j] = srow[hi * 8 + j];
        if (hi == 0) drow[16] = feats[ec];            // cols 64..67
    }
    __syncthreads();

    // WMMA main loop, fully unrolled: accN is the 16x16 tile at cols N*16..
    v8f acc0 = {}, acc1 = {}, acc2 = {}, acc3 = {};
    const float* arow = aw + row16 * DIN;             // + kb (even) -> b64 load
    const float* wbase = sW2 + row16 * 2;             // + pair*128 + nt*32
    #pragma unroll
    for (int k = 0; k < 17; ++k) {
        const int kb   = k * 4 + hi * 2;              // K index (even)
        const int pair = 2 * k + hi;                  // K-pair index
        v2f a  = *(const v2f*)(arow + kb);
        const float* wp = wbase + pair * (DSTATE * 2);
        v2f b0 = *(const v2f*)(wp +  0);
        v2f b1 = *(const v2f*)(wp + 32);
        v2f b2 = *(const v2f*)(wp + 64);
        v2f b3 = *(const v2f*)(wp + 96);
        acc0 = __builtin_amdgcn_wmma_f32_16x16x4_f32(false, a, false, b0, (short)0, acc0, false, false);
        acc1 = __builtin_amdgcn_wmma_f32_16x16x4_f32(false, a, false, b1, (short)0, acc1, false, false);
        acc2 = __builtin_amdgcn_wmma_f32_16x16x4_f32(false, a, false, b2, (short)0, acc2, false, false);
        acc3 = __builtin_amdgcn_wmma_f32_16x16x4_f32(false, a, false, b3, (short)0, acc3, false, false);
    }

    // Epilogue: bias + tanh + scatter atomic-add.
    // C layout: VGPR j -> M = hi*8 + j, N = nt*16 + row16.
    int  dstRow[8];
    bool valid[8];
    #pragma unroll
    for (int j = 0; j < 8; ++j) {
        long e  = tileBase + hi * 8 + j;
        valid[j]  = (e < E);
        long ec = valid[j] ? e : (long)(E - 1);       // unconditional clamped load
        dstRow[j] = eto[ec];
    }
    const float b0c = sB[ 0 + row16];
    const float b1c = sB[16 + row16];
    const float b2c = sB[32 + row16];
    const float b3c = sB[48 + row16];
    #pragma unroll
    for (int j = 0; j < 8; ++j) {
        if (valid[j]) {
            float* drow = state_next + (long)dstRow[j] * DSTATE + row16;
            atomicAdd(drow +  0, tanhf(acc0[j] + b0c));
            atomicAdd(drow + 16, tanhf(acc1[j] + b1c));
            atomicAdd(drow + 32, tanhf(acc2[j] + b2c));
            atomicAdd(drow + 48, tanhf(acc3[j] + b3c));
        }
    }
}

// ---------------------------------------------------------------------------
// Segment-sum: gacc[g, c] += state[n, c]   (state is L2 resident)
__global__ void gnn_seg_kernel(const float* __restrict__ state,
                               const int* __restrict__ gidx,
                               float* __restrict__ gacc, int N) {
    int i = blockIdx.x * blockDim.x + threadIdx.x;
    if (i >= N * DSTATE) return;
    int n = i >> 6, c = i & 63;
    atomicAdd(&gacc[(long)gidx[n] * DSTATE + c], state[i]);
}

// ---------------------------------------------------------------------------
// out[g, o] = b_out[o] + sum_c gacc[g, c] * W_out[c, o]
__global__ void gnn_out_kernel(const float* __restrict__ gacc,
                               const float* __restrict__ Wout,  // [64,2]
                               const float* __restrict__ bout,  // [2]
                               float* __restrict__ out, int G) {
    int i = blockIdx.x * blockDim.x + threadIdx.x;
    if (i >= G * 2) return;
    int g = i >> 1, o = i & 1;
    const float* gr = gacc + (long)g * DSTATE;
    float s = bout[o];
    #pragma unroll
    for (int c = 0; c < DSTATE; ++c) s += gr[c] * Wout[c * 2 + o];
    out[i] = s;
}

// ---------------------------------------------------------------------------
extern "C" void kernel_launch(void* const* d_in, const int* in_sizes, int n_in,
                              void* d_out, int out_size, void* d_ws, size_t ws_size,
                              hipStream_t stream) {
    const float* coords = (const float*)d_in[0];   // [N,3]
    const float* elen   = (const float*)d_in[1];   // [E,1]
    const float* evec   = (const float*)d_in[2];   // [E,3]
    const float* Wmsg   = (const float*)d_in[3];   // [68,64]
    const float* bmsg   = (const float*)d_in[4];   // [64]
    const float* Wout   = (const float*)d_in[5];   // [64,2]
    const float* bout   = (const float*)d_in[6];   // [2]
    const int*   gidx   = (const int*)d_in[7];     // [N]
    const int*   eto    = (const int*)d_in[8];     // [E]
    const int*   efrom  = (const int*)d_in[9];     // [E]

    const int N = in_sizes[0] / 3;
    const int E = in_sizes[1];
    const int G = out_size / 2;

    // Workspace layout (all f32):
    float* stateA = (float*)d_ws;                          // N*64
    float* stateB = stateA + (size_t)N * DSTATE;           // N*64
    float* featsW = stateB + (size_t)N * DSTATE;           // E*4
    float* gacc   = featsW + (size_t)E * 4;                // G*64

    const long nState = (long)N * DSTATE;

    // Zero initial state and the segment accumulator.
    {
        long tot = nState;
        gnn_zero_kernel<<<(int)((tot + 255) / 256), 256, 0, stream>>>(stateA, tot);
        long tg = (long)G * DSTATE;
        gnn_zero_kernel<<<(int)((tg + 255) / 256), 256, 0, stream>>>(gacc, tg);
    }

    // Edge-invariant features (loop invariant).
    gnn_feats_kernel<<<(E + 255) / 256, 256, 0, stream>>>(
        coords, elen, evec, efrom, eto, (float4*)featsW, E);

    // Message-passing rounds (reference: num_rounds = 3).
    const int tiles  = (E + 15) / 16;
    const int blocks = (tiles + WPB - 1) / WPB;
    float* cur = stateA;
    float* nxt = stateB;
    for (int r = 0; r < NUM_ROUNDS; ++r) {
        hipMemcpyAsync(nxt, cur, (size_t)nState * sizeof(float),
                       hipMemcpyDeviceToDevice, stream);
        gnn_msg_kernel<<<blocks, 256, 0, stream>>>(
            cur, nxt, (const float4*)featsW, Wmsg, bmsg, efrom, eto, E);
        float* t = cur; cur = nxt; nxt = t;
    }

    // Readout.
    gnn_seg_kernel<<<(int)(((long)N * DSTATE + 255) / 256), 256, 0, stream>>>(
        cur, gidx, gacc, N);
    gnn_out_kernel<<<(G * 2 + 255) / 256, 256, 0, stream>>>(
        gacc, Wout, bout, (float*)d_out, G);
}